// VocosVQCodec_56392920596581
// MI455X (gfx1250) — compile-verified
//
#include <hip/hip_runtime.h>

typedef float v2f __attribute__((ext_vector_type(2)));
typedef float v8f __attribute__((ext_vector_type(8)));

#define B_   32
#define D_   100
#define T_   2048
#define N_   65536          // B_*T_
#define K_   1024
#define NCB  4
#define WAVES 4
#define ROWS 64             // rows per workgroup (16 per wave)
#define BLOCK 128
#define KT   (K_ / 16)      // 64 k-tiles
#define CHUNKS (D_ / 4)     // 25 WMMA K-chunks
#define RS   104            // padded LDS row stride (floats, even -> 8B aligned pairs)

#define IDX_OFF    (B_ * D_ * T_)          // 6,553,600
#define COMMIT_OFF (IDX_OFF + NCB * N_)    // 6,815,744
#define UTIL_OFF   (COMMIT_OFF + 1)

// ---------------- workspace init: ws[0]=commit sum, ws[1..4096]=used flags ----
__global__ void rvq_init_ws(float* __restrict__ ws) {
  int i = blockIdx.x * blockDim.x + threadIdx.x;
  if (i < 1 + NCB * K_) ws[i] = 0.0f;
}

// ---------------- fused ResidualVQ kernel ------------------------------------
__global__ __launch_bounds__(BLOCK)
void rvq_main(const float* __restrict__ x, const float* __restrict__ embed,
              float* __restrict__ out, float* __restrict__ ws) {
  __shared__ float s_res[ROWS * RS];        // live residual, 64 rows
  __shared__ float s_e[16 * RS];            // staged e k-tile
  __shared__ float s_e2[16];                // ||e_k||^2 for the tile
  __shared__ float s_sc[WAVES][16][16];     // argmin reduction: scores
  __shared__ int   s_ix[WAVES][16][16];     // argmin reduction: indices
  __shared__ int   s_widx[WAVES][16];       // winning code per row
  __shared__ float s_com[BLOCK];            // commit-loss reduction

  const int tid  = threadIdx.x;
  const int lane = tid & 31;
  const int wave = tid >> 5;
  const int rowBase = blockIdx.x * ROWS;    // 64 | 2048 => block never straddles batch b

  // Load residual := x rows (coalesced: consecutive tid -> consecutive t)
  for (int i = tid; i < ROWS * D_; i += BLOCK) {
    int d = i / ROWS, r = i % ROWS;
    int n = rowBase + r;
    int b = n >> 11, t = n & (T_ - 1);
    s_res[r * RS + d] = x[(size_t)b * D_ * T_ + (size_t)d * T_ + t];
  }

  float commitAcc = 0.0f;
  const int mrow = lane & 15;               // A row / B col owned by this lane
  const int koff = (lane < 16) ? 0 : 2;     // 16x16x4 f32 A/B lane-half K offset
  const int waveRow = wave * 16;

  for (int cb = 0; cb < NCB; ++cb) {
    const float* e = embed + (size_t)cb * K_ * D_;
    float best[8]; int bidx[8];
#pragma unroll
    for (int v = 0; v < 8; ++v) { best[v] = 3.4e38f; bidx[v] = 0; }

    // Hoist the wave's A fragments (16 residual rows) into registers:
    // invariant across all 64 k-tiles of this codebook. 25 x v2f = 50 VGPRs.
    v2f afrag[CHUNKS];
#pragma unroll
    for (int ch = 0; ch < CHUNKS; ++ch)
      afrag[ch] = *(const v2f*)&s_res[(waveRow + mrow) * RS + ch * 4 + koff];

    for (int kt = 0; kt < KT; ++kt) {
      __syncthreads();                      // previous tile fully consumed
      for (int i = tid; i < 16 * D_; i += BLOCK) {
        int kk = i / D_, d = i % D_;
        s_e[kk * RS + d] = e[(size_t)(kt * 16 + kk) * D_ + d];
      }
      __syncthreads();
      if (tid < 16) {
        float s = 0.f;
        for (int d = 0; d < D_; ++d) { float v = s_e[tid * RS + d]; s += v * v; }
        s_e2[tid] = s;
      }
      __syncthreads();

      // G[16x16] = residual_tile(16xD) @ e_tile^T(Dx16); two independent
      // accumulation chains to halve dependent XDL issue depth.
      v8f c0 = {}, c1 = {};
#pragma unroll
      for (int ch = 0; ch < CHUNKS; ++ch) {
        v2f b = *(const v2f*)&s_e[mrow * RS + ch * 4 + koff];
        if (ch & 1)
          c1 = __builtin_amdgcn_wmma_f32_16x16x4_f32(
                   false, afrag[ch], false, b, (short)0, c1, false, false);
        else
          c0 = __builtin_amdgcn_wmma_f32_16x16x4_f32(
                   false, afrag[ch], false, b, (short)0, c0, false, false);
      }

      // score = ||e||^2 - 2*G  (x^2 term is row-constant, irrelevant to argmin)
      float e2v = s_e2[mrow];
      int kbase = kt * 16 + mrow;
#pragma unroll
      for (int v = 0; v < 8; ++v) {
        float score = e2v - 2.0f * (c0[v] + c1[v]); // C: M = v + 8*(lane>=16), N = lane&15
        if (score < best[v]) { best[v] = score; bidx[v] = kbase; }
      }
    }

    // Cross-lane argmin per row (first-lowest-index tie break)
#pragma unroll
    for (int v = 0; v < 8; ++v) {
      int row = v + ((lane < 16) ? 0 : 8);
      s_sc[wave][row][mrow] = best[v];
      s_ix[wave][row][mrow] = bidx[v];
    }
    __syncthreads();
    if (lane < 16) {
      float bs = 3.4e38f; int bi = 0x7fffffff;
      for (int cix = 0; cix < 16; ++cix) {
        float sc = s_sc[wave][lane][cix];
        int   kk = s_ix[wave][lane][cix];
        if (sc < bs || (sc == bs && kk < bi)) { bs = sc; bi = kk; }
      }
      s_widx[wave][lane] = bi;
      int n = rowBase + waveRow + lane;
      out[IDX_OFF + (size_t)cb * N_ + n] = (float)bi;   // indices as float
      ws[1 + cb * K_ + bi] = 1.0f;                      // utilization flag
    }
    __syncthreads();

    // residual -= e[idx]; commit += ||residual_new||^2 (q gathers hit L2)
    for (int i = lane; i < 16 * D_; i += 32) {
      int r = i / D_, d = i % D_;
      int kk = s_widx[wave][r];
      float qv = e[(size_t)kk * D_ + d];
      float rn = s_res[(waveRow + r) * RS + d] - qv;
      commitAcc += rn * rn;
      s_res[(waveRow + r) * RS + d] = rn;
    }
  }

  // q_total = x - residual_final  (coalesced store in [B,D,T] layout)
  __syncthreads();
  for (int i = tid; i < ROWS * D_; i += BLOCK) {
    int d = i / ROWS, r = i % ROWS;
    int n = rowBase + r;
    int b = n >> 11, t = n & (T_ - 1);
    size_t g = (size_t)b * D_ * T_ + (size_t)d * T_ + t;
    out[g] = x[g] - s_res[r * RS + d];
  }

  // block-reduce commitment loss -> one atomic per block
  s_com[tid] = commitAcc;
  __syncthreads();
  if (tid == 0) {
    float s = 0.f;
    for (int i = 0; i < BLOCK; ++i) s += s_com[i];
    atomicAdd(&ws[0], s);
  }
}

// ---------------- finalize: scalar outputs -----------------------------------
__global__ void rvq_finalize(const float* __restrict__ ws, float* __restrict__ out) {
  __shared__ float red[256];
  float s = 0.f;
  for (int i = threadIdx.x; i < NCB * K_; i += 256) s += ws[1 + i];
  red[threadIdx.x] = s;
  __syncthreads();
  if (threadIdx.x == 0) {
    float u = 0.f;
    for (int i = 0; i < 256; ++i) u += red[i];
    out[UTIL_OFF]   = u / (float)K_ / (float)NCB;
    out[COMMIT_OFF] = ws[0] / ((float)N_ * (float)D_ * (float)NCB);
  }
}

extern "C" void kernel_launch(void* const* d_in, const int* in_sizes, int n_in,
                              void* d_out, int out_size, void* d_ws, size_t ws_size,
                              hipStream_t stream) {
  (void)in_sizes; (void)n_in; (void)out_size; (void)ws_size;
  const float* x     = (const float*)d_in[0];   // [32, 100, 2048] f32
  const float* embed = (const float*)d_in[1];   // [4, 1024, 100] f32
  float* out = (float*)d_out;
  float* ws  = (float*)d_ws;

  rvq_init_ws<<<(1 + NCB * K_ + 255) / 256, 256, 0, stream>>>(ws);
  rvq_main<<<N_ / ROWS, BLOCK, 0, stream>>>(x, embed, out, ws);
  rvq_finalize<<<1, 256, 0, stream>>>(ws, out);
}